// AttnLayer_60636348285536
// MI455X (gfx1250) — compile-verified
//
#include <hip/hip_runtime.h>
#include <hip/hip_bf16.h>

typedef float v2f __attribute__((ext_vector_type(2)));
typedef float v8f __attribute__((ext_vector_type(8)));

#define Wd 14
#define Hd 14
#define Nf 3
#define WH 196           // Wd*Hd, divisible by 4 -> K-chunks never straddle t
#define Tt 64
#define ROWSTRIDE 12544  // Tt*WH, per-(b,c) row length in floats

// One block = one 16-row output tile. 8 waves split the T dimension (8 t each),
// every wave runs the f32 WMMA K-loop, results merged via LDS float atomics.
__global__ __launch_bounds__(256) void gauss_pool_wmma(
    const float* __restrict__ video,
    const float* __restrict__ mu_x, const float* __restrict__ mu_y,
    const float* __restrict__ sigma_x, const float* __restrict__ sigma_y,
    float* __restrict__ out)
{
    __shared__ float fLDS[16 * WH];   // B-matrix image: rows 0..2 = filters, 3..15 = 0
    __shared__ float accum[16 * 16];  // cross-wave C reduction
    __shared__ float scale[Nf];

    const int tid = threadIdx.x;

    if (tid < 256) accum[tid] = 0.0f;

    // ---- Phase 1: build normalized filter (1/T folded in) into LDS ----
    for (int idx = tid; idx < 16 * WH; idx += 256) {
        float val = 0.0f;
        if (idx < Nf * WH) {
            int n = idx / WH;
            int i = idx - n * WH;
            int w = i / Hd;
            int h = i - w * Hd;
            float mx  = tanhf(mu_x[n]);
            float my  = tanhf(mu_y[n]);
            float sgx = 1.0f / (1.0f + __expf(-sigma_x[n]));
            float sgy = 1.0f / (1.0f + __expf(-sigma_y[n]));
            float sx  = __expf(1.5f - 2.0f * sgx);
            float sy  = __expf(1.5f - 2.0f * sgy);
            float invx = 1.0f / (sx * sx + 1e-6f);
            float invy = 1.0f / (sy * sy + 1e-6f);
            float mux = (float)(Wd - 1) * (mx + 1.0f) * 0.5f;
            float muy = (float)(Hd - 1) * (my + 1.0f) * 0.5f;
            float dx = (float)w - mux;
            float dy = (float)h - muy;
            val = __expf(-0.5f * (dx * dx * invx + dy * dy * invy));
        }
        fLDS[idx] = val;
    }
    __syncthreads();

    if (tid < Nf) {
        float s = 0.0f;
        for (int i = 0; i < WH; ++i) s += fLDS[tid * WH + i];
        scale[tid] = 1.0f / ((s + 1e-6f) * (float)Tt);
    }
    __syncthreads();
    for (int idx = tid; idx < Nf * WH; idx += 256)
        fLDS[idx] *= scale[idx / WH];
    __syncthreads();

    // ---- Phase 2: streaming WMMA K-loop ----
    const int lane = tid & 31;
    const int wave = tid >> 5;      // 0..7 -> t-slice [wave*8, wave*8+8)
    const int hi   = lane >> 4;     // half-wave selector (K split per ISA layout)
    const int mn   = lane & 15;     // A row index == B column index for this lane

    const int tile = blockIdx.x;    // 512 tiles x 16 rows = 8192 = B*C
    const size_t rowBase = (size_t)(tile * 16 + mn) * ROWSTRIDE;
    const float* pA = video + rowBase + (size_t)wave * 8 * WH + 2 * hi;
    const float* pB = &fLDS[mn * WH + 2 * hi];

    v8f acc = {0.f, 0.f, 0.f, 0.f, 0.f, 0.f, 0.f, 0.f};

    for (int tt = 0; tt < 8; ++tt) {
        const float* pa = pA + tt * WH;
        #pragma unroll 7
        for (int c = 0; c < 49; ++c) {        // 49 K-chunks of 4 per t-slice
            v2f a = *(const v2f*)(pa + c * 4);   // global_load_b64, 8B aligned
            v2f b = *(const v2f*)(pB + c * 4);   // ds_load_b64, conflict-free
            acc = __builtin_amdgcn_wmma_f32_16x16x4_f32(
                false, a, false, b, (short)0, acc, false, false);
        }
    }

    // C/D layout: VGPR j, lanes 0-15 -> (M=j, N=lane); lanes 16-31 -> (M=j+8)
    #pragma unroll
    for (int j = 0; j < 8; ++j)
        atomicAdd(&accum[(j + 8 * hi) * 16 + mn], acc[j]);   // ds_add_f32
    __syncthreads();

    if (tid < 16 * Nf) {
        int m = tid / Nf;
        int n = tid - m * Nf;
        out[(size_t)(tile * 16 + m) * Nf + n] = accum[m * 16 + n];
    }
}

extern "C" void kernel_launch(void* const* d_in, const int* in_sizes, int n_in,
                              void* d_out, int out_size, void* d_ws, size_t ws_size,
                              hipStream_t stream) {
    (void)in_sizes; (void)n_in; (void)d_ws; (void)ws_size; (void)out_size;
    const float* video   = (const float*)d_in[0];
    const float* mu_x    = (const float*)d_in[1];
    const float* mu_y    = (const float*)d_in[2];
    const float* sigma_x = (const float*)d_in[3];
    const float* sigma_y = (const float*)d_in[4];
    float* out = (float*)d_out;

    dim3 grid(512);   // 8192 (b,c) rows / 16 rows per tile
    dim3 block(256);  // 8 wave32s per tile
    gauss_pool_wmma<<<grid, block, 0, stream>>>(video, mu_x, mu_y,
                                                sigma_x, sigma_y, out);
}